// Word2Vec_27393301413987
// MI455X (gfx1250) — compile-verified
//
#include <hip/hip_runtime.h>

// Word2Vec negative-sampling forward:
//   out[i,k] = dot(W_out[oidx[i,k], :], W_in[:, iidx[i]])   (B=16384, K=64, D=256)
//
// One wave32 per batch element. f32 WMMA 16x16x4 with the context vector u
// broadcast across the 16 B-columns (exact f32 math; kernel is gather-bound so
// the 15/16 lane waste is free). 4 accumulator tiles cover K=64.

typedef __attribute__((ext_vector_type(2))) float v2f;
typedef __attribute__((ext_vector_type(8))) float v8f;

#define WVDIM 256           // D
#define KNEG 64             // K
#define WAVES_PER_BLOCK 8   // 256 threads/block

__global__ __launch_bounds__(256) void w2v_wmma_f32_kernel(
    const float* __restrict__ W_in,      // [D, num_tokens]
    const float* __restrict__ W_out,     // [num_tokens, D]
    const int*   __restrict__ in_idx,    // [B]
    const int*   __restrict__ out_idx,   // [B, K]
    float*       __restrict__ out,       // [B, K]
    int num_tokens, int batch)
{
    __shared__ float u_lds[WAVES_PER_BLOCK][WVDIM];

    const int lane = threadIdx.x & 31;
    const int wave = threadIdx.x >> 5;
    const int i    = blockIdx.x * WAVES_PER_BLOCK + wave;   // batch element
    if (i >= batch) return;                                  // wave-uniform

    // ---- Gather u = W_in[:, in_idx[i]] into LDS (column gather, stride NT) ----
    const int ii = in_idx[i];
    const float* Wi = W_in + ii;
    #pragma unroll
    for (int j = 0; j < WVDIM / 32; ++j) {
        u_lds[wave][lane + 32 * j] = Wi[(size_t)(lane + 32 * j) * num_tokens];
    }
    __syncthreads();

    const int hi  = lane >> 4;       // 0: lanes 0-15 (wmma-K 0,1), 1: lanes 16-31 (wmma-K 2,3)
    const int hi2 = hi * 2;
    const int m   = lane & 15;       // row within 16x16 tile

    // ---- A-operand row pointers: 4 tiles x 16 rows of W_out ----
    const float* vp0; const float* vp1; const float* vp2; const float* vp3;
    {
        const int* oi = out_idx + (size_t)i * KNEG + m;
        vp0 = W_out + (size_t)oi[0]  * WVDIM + hi2;
        vp1 = W_out + (size_t)oi[16] * WVDIM + hi2;
        vp2 = W_out + (size_t)oi[32] * WVDIM + hi2;
        vp3 = W_out + (size_t)oi[48] * WVDIM + hi2;
    }

    v8f acc0 = {}, acc1 = {}, acc2 = {}, acc3 = {};
    const float* up = &u_lds[wave][hi2];

    // ---- Reduction over D in WMMA-K=4 chunks; 4 independent accumulators ----
    #pragma unroll 8
    for (int s = 0; s < WVDIM / 4; ++s) {
        // B 4x16 f32, all columns == u[d0..d0+3]; per-lane pair per K-split.
        v2f bv = *(const v2f*)(up + 4 * s);        // ds_load_b64, half-wave broadcast
        v2f a0 = *(const v2f*)(vp0 + 4 * s);       // global_load_b64 each
        v2f a1 = *(const v2f*)(vp1 + 4 * s);
        v2f a2 = *(const v2f*)(vp2 + 4 * s);
        v2f a3 = *(const v2f*)(vp3 + 4 * s);
        acc0 = __builtin_amdgcn_wmma_f32_16x16x4_f32(false, a0, false, bv, (short)0, acc0, false, false);
        acc1 = __builtin_amdgcn_wmma_f32_16x16x4_f32(false, a1, false, bv, (short)0, acc1, false, false);
        acc2 = __builtin_amdgcn_wmma_f32_16x16x4_f32(false, a2, false, bv, (short)0, acc2, false, false);
        acc3 = __builtin_amdgcn_wmma_f32_16x16x4_f32(false, a3, false, bv, (short)0, acc3, false, false);
    }

    // ---- D 16x16: VGPR r, lane-half h -> M = r + 8h; take column N==0 ----
    if (m == 0) {
        float* op = out + (size_t)i * KNEG + hi * 8;
        #pragma unroll
        for (int r = 0; r < 8; ++r) {
            op[r]      = acc0[r];
            op[16 + r] = acc1[r];
            op[32 + r] = acc2[r];
            op[48 + r] = acc3[r];
        }
    }
}

extern "C" void kernel_launch(void* const* d_in, const int* in_sizes, int n_in,
                              void* d_out, int out_size, void* d_ws, size_t ws_size,
                              hipStream_t stream) {
    const float* W_in   = (const float*)d_in[0];   // [D, num_tokens]
    const float* W_out  = (const float*)d_in[1];   // [num_tokens, D]
    const int*   in_idx = (const int*)d_in[2];     // [B]
    const int*   out_idx= (const int*)d_in[3];     // [B, K]
    float* out = (float*)d_out;

    const int batch      = in_sizes[2];
    const int num_tokens = in_sizes[0] / WVDIM;

    const int blocks = (batch + WAVES_PER_BLOCK - 1) / WAVES_PER_BLOCK;   // 2048
    w2v_wmma_f32_kernel<<<blocks, WAVES_PER_BLOCK * 32, 0, stream>>>(
        W_in, W_out, in_idx, out_idx, out, num_tokens, batch);
}